// MetaJanusAttention_50440095924326
// MI455X (gfx1250) — compile-verified
//
#include <hip/hip_runtime.h>

// MI455X / gfx1250, wave32. FP32 via V_WMMA_F32_16X16X4_F32 (workload is
// HBM-bound: ~110MB unique traffic vs ~70 GFLOP, so fp32 WMMA is free).
// GEMM staging uses GLOBAL_LOAD_ASYNC_TO_LDS_B128 + double-buffered LDS,
// tracked with ASYNCcnt (s_wait_asynccnt), per CDNA5 ISA.

typedef __attribute__((ext_vector_type(2))) float v2f;
typedef __attribute__((ext_vector_type(8))) float v8f;

#define BDIM 4
#define TDIM 2048
#define EDIM 1024
#define MROWS (BDIM * TDIM)  // 8192

// --- CDNA5 async copy: global -> LDS, 16 bytes per lane, ASYNCcnt-tracked ---
__device__ __forceinline__ void async_cp_b128(const float* g, const float* lds) {
    unsigned lo = (unsigned)(size_t)lds;  // flat shared addr low 32 bits == LDS byte offset
    asm volatile("global_load_async_to_lds_b128 %0, %1, off"
                 :: "v"(lo), "v"(g) : "memory");
}
__device__ __forceinline__ void wait_async_le6() {
    asm volatile("s_wait_asynccnt 0x6" ::: "memory");
}
__device__ __forceinline__ void wait_async_le0() {
    asm volatile("s_wait_asynccnt 0x0" ::: "memory");
}

// ---------------------------------------------------------------------------
// C[M,N] = A[M,K] * W[N,K]^T   (all fp32, row-major)
// Block tile 64M x 128N, K-panel 32, 8 waves (2x4), wave tile 32x32
// = 2x2 WMMA 16x16 accumulators, fp32 WMMA K-step 4.
// Panels staged via async-to-LDS into double buffers: compute panel p from
// buffer p&1 while the async engine fills buffer (p+1)&1.
// ---------------------------------------------------------------------------
__global__ __launch_bounds__(256) void gemm_nt(const float* __restrict__ A,
                                               const float* __restrict__ W,
                                               float* __restrict__ C,
                                               int M, int N, int K) {
    __shared__ __align__(16) float As[2][64][36];   // pad 36: b128 stores, v2f reads
    __shared__ __align__(16) float Ws[2][128][36];

    const int tid  = threadIdx.x;
    const int lane = tid & 31;
    const int wave = tid >> 5;
    const int wm   = wave >> 2;     // 0..1  (M dir)
    const int wn   = wave & 3;      // 0..3  (N dir)
    const int half = lane >> 4;     // lane half selects K pair / row offset
    const int l16  = lane & 15;

    const int m0 = blockIdx.x * 64;
    const int n0 = blockIdx.y * 128;

    v8f acc[2][2] = {};

    const int npan = K >> 5;

    // Prologue: kick off panel 0 into buffer 0 (6 async b128 per thread).
    {
#pragma unroll
        for (int i = 0; i < 2; ++i) {
            int e = (tid + i * 256) << 2;
            int row = e >> 5, col = e & 31;
            async_cp_b128(A + (size_t)(m0 + row) * K + col, &As[0][row][col]);
        }
#pragma unroll
        for (int i = 0; i < 4; ++i) {
            int e = (tid + i * 256) << 2;
            int row = e >> 5, col = e & 31;
            async_cp_b128(W + (size_t)(n0 + row) * K + col, &Ws[0][row][col]);
        }
    }

    for (int p = 0; p < npan; ++p) {
        const int cur = p & 1;
        // All waves done reading the buffer we are about to overwrite.
        __syncthreads();
        if (p + 1 < npan) {
            const int kn = (p + 1) << 5;
#pragma unroll
            for (int i = 0; i < 2; ++i) {
                int e = (tid + i * 256) << 2;
                int row = e >> 5, col = e & 31;
                async_cp_b128(A + (size_t)(m0 + row) * K + kn + col, &As[cur ^ 1][row][col]);
            }
#pragma unroll
            for (int i = 0; i < 4; ++i) {
                int e = (tid + i * 256) << 2;
                int row = e >> 5, col = e & 31;
                async_cp_b128(W + (size_t)(n0 + row) * K + kn + col, &Ws[cur ^ 1][row][col]);
            }
            // Async loads retire in order: <=6 outstanding => panel p has landed.
            wait_async_le6();
        } else {
            wait_async_le0();
        }
        if (p + 2 < npan) {   // L2 prefetch two panels ahead (global_prefetch_b8)
            const int kf = (p + 2) << 5;
            __builtin_prefetch(A + (size_t)(m0 + (tid & 63)) * K + kf, 0, 1);
            __builtin_prefetch(W + (size_t)(n0 + (tid & 127)) * K + kf, 0, 1);
        }
        __syncthreads();   // publish panel p across all waves

#pragma unroll
        for (int kk = 0; kk < 32; kk += 4) {
            // A frag (16x4 f32): v0 = K+2*half, v1 = K+1+2*half, row = l16
            v2f a0 = *reinterpret_cast<const v2f*>(&As[cur][wm * 32 +      l16][kk + 2 * half]);
            v2f a1 = *reinterpret_cast<const v2f*>(&As[cur][wm * 32 + 16 + l16][kk + 2 * half]);
            // B frag (4x16 f32): Bᵀ == W rows; col = l16, K pair by lane half
            v2f b0 = *reinterpret_cast<const v2f*>(&Ws[cur][wn * 32 +      l16][kk + 2 * half]);
            v2f b1 = *reinterpret_cast<const v2f*>(&Ws[cur][wn * 32 + 16 + l16][kk + 2 * half]);

            acc[0][0] = __builtin_amdgcn_wmma_f32_16x16x4_f32(false, a0, false, b0, (short)0, acc[0][0], false, false);
            acc[0][1] = __builtin_amdgcn_wmma_f32_16x16x4_f32(false, a0, false, b1, (short)0, acc[0][1], false, false);
            acc[1][0] = __builtin_amdgcn_wmma_f32_16x16x4_f32(false, a1, false, b0, (short)0, acc[1][0], false, false);
            acc[1][1] = __builtin_amdgcn_wmma_f32_16x16x4_f32(false, a1, false, b1, (short)0, acc[1][1], false, false);
        }
    }

    // C/D layout: VGPR r -> row r (lanes 0-15) or r+8 (lanes 16-31), col = l16
#pragma unroll
    for (int mt = 0; mt < 2; ++mt)
#pragma unroll
        for (int nt = 0; nt < 2; ++nt) {
            const int mbase = m0 + wm * 32 + mt * 16 + 8 * half;
            const int n     = n0 + wn * 32 + nt * 16 + l16;
#pragma unroll
            for (int r = 0; r < 8; ++r)
                C[(size_t)(mbase + r) * N + n] = acc[mt][nt][r];
        }
}

// ---------------------------------------------------------------------------
// scores[i] = ||J_i||^2 / sqrt(E)   (identity: x.(J Wj) row-dot == ||J||^2,
// which deletes the entire echo-back GEMM)
// ---------------------------------------------------------------------------
__global__ __launch_bounds__(256) void row_sumsq(const float* __restrict__ J,
                                                 float* __restrict__ scores) {
    __shared__ float red[256];
    const int row = blockIdx.x;
    const float* p = J + (size_t)row * EDIM;
    float s = 0.f;
    for (int j = threadIdx.x; j < EDIM; j += 256) {
        float v = p[j];
        s += v * v;
    }
    red[threadIdx.x] = s;
    __syncthreads();
    for (int off = 128; off > 0; off >>= 1) {
        if (threadIdx.x < off) red[threadIdx.x] += red[threadIdx.x + off];
        __syncthreads();
    }
    if (threadIdx.x == 0) scores[row] = red[0] * 0.03125f;  // 1/sqrt(1024)
}

// ---------------------------------------------------------------------------
// Per-row softmax stats over rank-1 causal logits: m_t, 1/Z_t.
// One wave32 per row; shuffle reductions.
// ---------------------------------------------------------------------------
__global__ __launch_bounds__(256) void softmax_stats(const float* __restrict__ sc,
                                                     float* __restrict__ mrow,
                                                     float* __restrict__ invz) {
    const int wave = threadIdx.x >> 5;
    const int lane = threadIdx.x & 31;
    const int t = blockIdx.x * 8 + wave;
    const int b = blockIdx.y;
    const float* a = sc + (size_t)b * TDIM;
    const float at = a[t];

    float mx = -3.0e38f;
    for (int s = lane; s <= t; s += 32) mx = fmaxf(mx, at * a[s]);
    for (int o = 16; o > 0; o >>= 1) mx = fmaxf(mx, __shfl_xor(mx, o, 32));

    float sum = 0.f;
    for (int s = lane; s <= t; s += 32) sum += expf(at * a[s] - mx);
    for (int o = 16; o > 0; o >>= 1) sum += __shfl_xor(sum, o, 32);

    if (lane == 0) {
        mrow[b * TDIM + t] = mx;
        invz[b * TDIM + t] = 1.0f / sum;
    }
}

// ---------------------------------------------------------------------------
// O[b,t,:] = sum_{s<=t} softmax_s(a_t a_s) * V[b,s,:]
// Workgroup: 16 rows x 256 cols. 8 waves, each 16x32 (2 WMMA N-tiles).
// P tile (16x16) built in LDS from rank-1 logits; V block staged transposed
// so B-fragments are single ds_load_b64. V (33MB) lives in the 192MB L2.
// ---------------------------------------------------------------------------
__global__ __launch_bounds__(256) void rank1_attention(const float* __restrict__ V,
                                                       const float* __restrict__ sc,
                                                       const float* __restrict__ mrow,
                                                       const float* __restrict__ invz,
                                                       float* __restrict__ O) {
    __shared__ float VT[256][18];   // [n][k], pad 18 keeps v2f even-aligned
    __shared__ float P[16][18];     // [t][s]
    __shared__ float sAt[16], sM[16], sIZ[16];

    const int tid  = threadIdx.x;
    const int lane = tid & 31;
    const int wave = tid >> 5;
    const int half = lane >> 4;
    const int l16  = lane & 15;

    const int t0 = blockIdx.x * 16;
    const int n0 = blockIdx.y * 256;
    const int b  = blockIdx.z;
    const size_t vbase = (size_t)b * TDIM * EDIM;
    const int rbase = b * TDIM + t0;

    if (tid < 16) {
        sAt[tid] = sc[rbase + tid];
        sM[tid]  = mrow[rbase + tid];
        sIZ[tid] = invz[rbase + tid];
    }

    v8f acc[2] = {};
    const int pr = tid >> 4;   // P row (t)
    const int pc = tid & 15;   // P col (s)

    for (int s0 = 0; s0 <= t0; s0 += 16) {
        __syncthreads();
        // Stage V[s0..s0+15, n0..n0+255] transposed into VT[n][k]
        {
            const float* src = V + vbase + (size_t)s0 * EDIM + n0 + tid;
#pragma unroll
            for (int k = 0; k < 16; ++k) VT[tid][k] = src[(size_t)k * EDIM];
        }
        // Probability tile: P[t][s] = exp(a_t a_s - m_t) / Z_t, causal mask
        {
            const int s = s0 + pc;
            const int t = t0 + pr;
            const float l = sAt[pr] * sc[b * TDIM + s];
            P[pr][pc] = (s <= t) ? expf(l - sM[pr]) * sIZ[pr] : 0.0f;
        }
        __syncthreads();

#pragma unroll
        for (int kk = 0; kk < 16; kk += 4) {
            v2f af = *reinterpret_cast<const v2f*>(&P[l16][kk + 2 * half]);
            v2f b0 = *reinterpret_cast<const v2f*>(&VT[wave * 32 +      l16][kk + 2 * half]);
            v2f b1 = *reinterpret_cast<const v2f*>(&VT[wave * 32 + 16 + l16][kk + 2 * half]);
            acc[0] = __builtin_amdgcn_wmma_f32_16x16x4_f32(false, af, false, b0, (short)0, acc[0], false, false);
            acc[1] = __builtin_amdgcn_wmma_f32_16x16x4_f32(false, af, false, b1, (short)0, acc[1], false, false);
        }
    }

    float* dst = O + vbase;
#pragma unroll
    for (int nt = 0; nt < 2; ++nt)
#pragma unroll
        for (int r = 0; r < 8; ++r)
            dst[(size_t)(t0 + r + 8 * half) * EDIM + (n0 + wave * 32 + nt * 16 + l16)] = acc[nt][r];
}

// ---------------------------------------------------------------------------
// Host-side orchestration (graph-capture safe: launches only).
// ---------------------------------------------------------------------------
extern "C" void kernel_launch(void* const* d_in, const int* in_sizes, int n_in,
                              void* d_out, int out_size, void* d_ws, size_t ws_size,
                              hipStream_t stream) {
    const float* x  = (const float*)d_in[0];
    const float* Wj = (const float*)d_in[1];
    const float* Wv = (const float*)d_in[2];
    const float* Wo = (const float*)d_in[3];
    float* out = (float*)d_out;

    char* ws = (char*)d_ws;
    float* J   = (float*)(ws);                          // 33.5MB; later reused as attn output
    float* Vv  = (float*)(ws + 33554432);               // 33.5MB
    float* sc  = (float*)(ws + 67108864);               // 32KB  scores
    float* mr  = (float*)(ws + 67108864 + 32768);       // 32KB  row max
    float* iz  = (float*)(ws + 67108864 + 65536);       // 32KB  1/Z

    dim3 gemmGrid(MROWS / 64, EDIM / 128);

    // 1) J = x @ Wj^T
    gemm_nt<<<gemmGrid, 256, 0, stream>>>(x, Wj, J, MROWS, EDIM, EDIM);
    // 2) scores = ||J_i||^2 / 32   (replaces the echo-back GEMM entirely)
    row_sumsq<<<dim3(MROWS), 256, 0, stream>>>(J, sc);
    // 3) softmax stats m_t, 1/Z_t over causal rank-1 logits
    softmax_stats<<<dim3(TDIM / 8, BDIM), 256, 0, stream>>>(sc, mr, iz);
    // 4) V = x @ Wv^T
    gemm_nt<<<gemmGrid, 256, 0, stream>>>(x, Wv, Vv, MROWS, EDIM, EDIM);
    // 5) attnOut = P @ V   (writes into J buffer — J is dead now)
    rank1_attention<<<dim3(TDIM / 16, EDIM / 256, BDIM), 256, 0, stream>>>(Vv, sc, mr, iz, J);
    // 6) out = attnOut @ Wo^T
    gemm_nt<<<gemmGrid, 256, 0, stream>>>(J, Wo, out, MROWS, EDIM, EDIM);
}